// TemporalGraphAttention_2774548873490
// MI455X (gfx1250) — compile-verified
//
#include <hip/hip_runtime.h>
#include <math.h>

#define N_NODES 50000
#define N_EDGES 800000
#define DIM     256
#define NH      8
#define HD      32

typedef __attribute__((ext_vector_type(2))) float v2f;
typedef __attribute__((ext_vector_type(8))) float v8f;

// -------------------------------------------------------------------------
// init: zero aggregation buffer, seed per-head max = -inf, sum = 0
// -------------------------------------------------------------------------
__global__ void tga_init_kernel(float* __restrict__ agg,
                                float* __restrict__ hmax,
                                float* __restrict__ hsum) {
    size_t t      = (size_t)blockIdx.x * blockDim.x + threadIdx.x;
    size_t stride = (size_t)gridDim.x * blockDim.x;
    size_t total  = (size_t)N_NODES * DIM;
    for (size_t i = t; i < total; i += stride) agg[i] = 0.0f;
    if (t < NH) { hmax[t] = -__builtin_inff(); hsum[t] = 0.0f; }
}

// -------------------------------------------------------------------------
// two 16x16 output tiles (rows [row0,row0+16) and [row0+16,row0+32)) per wave,
// sharing each B fragment -> 1 global_load_b32 pair per 2 WMMAs, and two
// independent accumulator chains so the XDL pipe issues back-to-back.
// Epilogue split on the *uniform* full-block condition so 1562/1563 blocks
// take a branch-free store path (no per-store exec-mask churn).
// -------------------------------------------------------------------------
__device__ __forceinline__ void wmma_tile_gemm2(
    const float* __restrict__ As,        // LDS 32x256 slab (zero-padded tail)
    const float* __restrict__ W,         // [256,256] row-major
    const float* __restrict__ bias,      // [256]
    float* __restrict__ C,               // [M,256] row-major
    int row0, int M, int n0, int lm, int lh)
{
    v8f acc0 = {}, acc1 = {};
#pragma unroll 4
    for (int kb = 0; kb < DIM; kb += 4) {
        const int kk = kb + 2 * lh;      // ISA f32 16x4 A layout: lanes 16-31 hold K+2
        v2f a0; a0.x = As[lm * DIM + kk];
                a0.y = As[lm * DIM + kk + 1];
        v2f a1; a1.x = As[(16 + lm) * DIM + kk];
                a1.y = As[(16 + lm) * DIM + kk + 1];
        v2f b;  b.x  = W[(size_t)kk * DIM + n0 + lm];
                b.y  = W[(size_t)(kk + 1) * DIM + n0 + lm];
        acc0 = __builtin_amdgcn_wmma_f32_16x16x4_f32(
            false, a0, false, b, (short)0, acc0, false, false);
        acc1 = __builtin_amdgcn_wmma_f32_16x16x4_f32(
            false, a1, false, b, (short)0, acc1, false, false);
    }
    const float bb = bias[n0 + lm];
    // D layout: vgpr j -> M = j + 8*lh, N = lm
    float* __restrict__ Cp = C + (size_t)(row0 + 8 * lh) * DIM + n0 + lm;
    if (row0 + 32 <= M) {                // uniform: full block, branch-free stores
#pragma unroll
        for (int j = 0; j < 8; ++j) {
            Cp[(size_t)j * DIM]        = acc0[j] + bb;
            Cp[(size_t)(j + 16) * DIM] = acc1[j] + bb;
        }
    } else {                             // single tail block only
#pragma unroll
        for (int j = 0; j < 8; ++j) {
            const int r0 = row0 + j + 8 * lh;   // always < M (M % 16 == 0)
            Cp[(size_t)j * DIM] = acc0[j] + bb;
            if (r0 + 16 < M) Cp[(size_t)(j + 16) * DIM] = acc1[j] + bb;
        }
    }
}

// cooperative 32x256 slab load, float4 (b128) traffic, zero-pad past M
__device__ __forceinline__ void load_slab32(
    float* __restrict__ As, const float* __restrict__ A, int row0, int M)
{
    const float4* __restrict__ srcp = (const float4*)A;
    float4* __restrict__ dst4 = (float4*)As;
#pragma unroll
    for (int ii = 0; ii < 4; ++ii) {             // 2048 float4 / 512 threads
        const int i  = threadIdx.x + ii * 512;
        const int gr = row0 + (i >> 6);          // 64 float4 per row
        dst4[i] = (gr < M) ? srcp[(size_t)gr * 64 + (i & 63)]
                           : make_float4(0.f, 0.f, 0.f, 0.f);
    }
}

// -------------------------------------------------------------------------
// fused QKV GEMM: q/k/v = x @ W{q,k,v} + b{q,k,v}
// block = 512 threads = 16 waves; block owns 32 rows, wave w owns cols [16w,16w+16)
// -------------------------------------------------------------------------
__global__ __launch_bounds__(512) void tga_qkv_gemm(
    const float* __restrict__ A,
    const float* __restrict__ Wq, const float* __restrict__ bq, float* __restrict__ q,
    const float* __restrict__ Wk, const float* __restrict__ bk, float* __restrict__ k,
    const float* __restrict__ Wv, const float* __restrict__ bv, float* __restrict__ v)
{
    __shared__ float As[32 * DIM];       // 32 KB of 320 KB/WGP
    const int row0 = blockIdx.x * 32;
    load_slab32(As, A, row0, N_NODES);
    __syncthreads();

    const int wave = threadIdx.x >> 5;
    const int lane = threadIdx.x & 31;
    const int n0   = wave * 16;
    const int lm   = lane & 15;
    const int lh   = lane >> 4;

    wmma_tile_gemm2(As, Wq, bq, q, row0, N_NODES, n0, lm, lh);
    wmma_tile_gemm2(As, Wk, bk, k, row0, N_NODES, n0, lm, lh);
    wmma_tile_gemm2(As, Wv, bv, v, row0, N_NODES, n0, lm, lh);
}

// single-matrix variant for the output projection
__global__ __launch_bounds__(512) void tga_out_gemm(
    const float* __restrict__ A,
    const float* __restrict__ Wo, const float* __restrict__ bo, float* __restrict__ C)
{
    __shared__ float As[32 * DIM];
    const int row0 = blockIdx.x * 32;
    load_slab32(As, A, row0, N_NODES);
    __syncthreads();

    const int wave = threadIdx.x >> 5;
    const int lane = threadIdx.x & 31;
    wmma_tile_gemm2(As, Wo, bo, C, row0, N_NODES, wave * 16, lane & 15, lane >> 4);
}

// -------------------------------------------------------------------------
// signed float atomic max via int-max / uint-min bit trick
// -------------------------------------------------------------------------
__device__ __forceinline__ void atomicMaxFloat(float* addr, float val) {
    if (!signbit(val)) atomicMax((int*)addr, __float_as_int(val));
    else               atomicMin((unsigned int*)addr, __float_as_uint(val));
}

// -------------------------------------------------------------------------
// scores[e][h] = dot(q[dst],k[src])/sqrt(32) + temporal; block-reduce max/head
// thread = (edge, head); q/k gathers are L2-resident (51.2 MB each < 192 MB L2)
// -------------------------------------------------------------------------
__global__ __launch_bounds__(256) void tga_scores_kernel(
    const float* __restrict__ q, const float* __restrict__ k,
    const int* __restrict__ src, const int* __restrict__ dst,
    const float* __restrict__ edge_time, const float* __restrict__ node_time,
    const float* __restrict__ Wt, const float* __restrict__ bt,
    float* __restrict__ sc, float* __restrict__ hmax)
{
    const int tid = blockIdx.x * blockDim.x + threadIdx.x;
    const int e   = tid >> 3;
    const int h   = tid & 7;
    float s = -__builtin_inff();
    if (e < N_EDGES) {
        const int d  = dst[e];
        const int sn = src[e];
        const float td = edge_time[e] - node_time[d];
        const float sg = (td > 0.0f) ? 1.0f : ((td < 0.0f) ? -1.0f : 0.0f);
        const float lg = log1pf(fabsf(td) * (1.0f / 3600.0f));
        const float tw = sg * Wt[h] + lg * Wt[NH + h] + bt[h];
        const float4* qp = (const float4*)(q + (size_t)d  * DIM + h * HD);
        const float4* kp = (const float4*)(k + (size_t)sn * DIM + h * HD);
        float dot = 0.0f;
#pragma unroll
        for (int i = 0; i < HD / 4; ++i) {
            const float4 qv = qp[i], kv = kp[i];
            dot = fmaf(qv.x, kv.x, dot);
            dot = fmaf(qv.y, kv.y, dot);
            dot = fmaf(qv.z, kv.z, dot);
            dot = fmaf(qv.w, kv.w, dot);
        }
        s = dot * 0.17677669529663689f + tw;  // 32^-0.5
        sc[(size_t)e * NH + h] = s;
    }
    __shared__ float red[256];
    red[threadIdx.x] = s;
    __syncthreads();
    for (int off = 128; off >= 8; off >>= 1) {
        if (threadIdx.x < (unsigned)off)
            red[threadIdx.x] = fmaxf(red[threadIdx.x], red[threadIdx.x + off]);
        __syncthreads();
    }
    if (threadIdx.x < NH) atomicMaxFloat(&hmax[threadIdx.x], red[threadIdx.x]);
}

// -------------------------------------------------------------------------
// sc[e][h] <- exp(sc - hmax[h]); block-reduce sum -> hsum[h]
// -------------------------------------------------------------------------
__global__ __launch_bounds__(256) void tga_sumexp_kernel(
    float* __restrict__ sc, const float* __restrict__ hmax, float* __restrict__ hsum)
{
    const int tid = blockIdx.x * blockDim.x + threadIdx.x;
    const int e   = tid >> 3;
    const int h   = tid & 7;
    float ex = 0.0f;
    if (e < N_EDGES) {
        ex = expf(sc[(size_t)e * NH + h] - hmax[h]);
        sc[(size_t)e * NH + h] = ex;
    }
    __shared__ float red[256];
    red[threadIdx.x] = ex;
    __syncthreads();
    for (int off = 128; off >= 8; off >>= 1) {
        if (threadIdx.x < (unsigned)off)
            red[threadIdx.x] += red[threadIdx.x + off];
        __syncthreads();
    }
    if (threadIdx.x < NH) unsafeAtomicAdd(&hsum[threadIdx.x], red[threadIdx.x]);
}

// -------------------------------------------------------------------------
// agg[dst] += (exp/sum) * v[src]; thread = (edge, channel) -> coalesced
// hardware global_atomic_add_f32 into the L2-resident 51.2 MB accumulator
// -------------------------------------------------------------------------
__global__ __launch_bounds__(256) void tga_aggregate_kernel(
    const float* __restrict__ v, const float* __restrict__ sc,
    const float* __restrict__ hsum,
    const int* __restrict__ src, const int* __restrict__ dst,
    float* __restrict__ agg)
{
    const long long t = (long long)blockIdx.x * blockDim.x + threadIdx.x;
    if (t >= (long long)N_EDGES * DIM) return;
    const int e = (int)(t >> 8);
    const int c = (int)(t & 255);
    const int h = c >> 5;
    const float alpha = sc[(size_t)e * NH + h] / hsum[h];
    const float val   = alpha * v[(size_t)src[e] * DIM + c];
    unsafeAtomicAdd(&agg[(size_t)dst[e] * DIM + c], val);
}

// -------------------------------------------------------------------------
extern "C" void kernel_launch(void* const* d_in, const int* in_sizes, int n_in,
                              void* d_out, int out_size, void* d_ws, size_t ws_size,
                              hipStream_t stream) {
    const float* x         = (const float*)d_in[0];
    const int*   ei        = (const int*)d_in[1];   // [2,E]: src row, dst row
    const float* edge_time = (const float*)d_in[2];
    const float* node_time = (const float*)d_in[3];
    const float* Wq = (const float*)d_in[4];  const float* bq = (const float*)d_in[5];
    const float* Wk = (const float*)d_in[6];  const float* bk = (const float*)d_in[7];
    const float* Wv = (const float*)d_in[8];  const float* bv = (const float*)d_in[9];
    const float* Wt = (const float*)d_in[10]; const float* bt = (const float*)d_in[11];
    const float* Wo = (const float*)d_in[12]; const float* bo = (const float*)d_in[13];
    float* out = (float*)d_out;

    const int* src = ei;
    const int* dst = ei + N_EDGES;

    // workspace layout (floats): q,k,v,agg (N*256 each), sc (E*8), hmax/hsum (8 each)
    float* q    = (float*)d_ws;
    float* k    = q   + (size_t)N_NODES * DIM;
    float* v    = k   + (size_t)N_NODES * DIM;
    float* agg  = v   + (size_t)N_NODES * DIM;
    float* sc   = agg + (size_t)N_NODES * DIM;
    float* hmax = sc  + (size_t)N_EDGES * NH;
    float* hsum = hmax + NH;

    const int gemm_blocks = (N_NODES + 31) / 32;   // 1563, last block half-valid

    tga_init_kernel<<<4096, 256, 0, stream>>>(agg, hmax, hsum);

    tga_qkv_gemm<<<gemm_blocks, 512, 0, stream>>>(
        x, Wq, bq, q, Wk, bk, k, Wv, bv, v);

    const int eh_blocks = (N_EDGES * NH + 255) / 256;
    tga_scores_kernel<<<eh_blocks, 256, 0, stream>>>(
        q, k, src, dst, edge_time, node_time, Wt, bt, sc, hmax);

    tga_sumexp_kernel<<<eh_blocks, 256, 0, stream>>>(sc, hmax, hsum);

    const long long agg_threads = (long long)N_EDGES * DIM;
    tga_aggregate_kernel<<<(int)((agg_threads + 255) / 256), 256, 0, stream>>>(
        v, sc, hsum, src, dst, agg);

    tga_out_gemm<<<gemm_blocks, 512, 0, stream>>>(agg, Wo, bo, out);
}